// Attention_22325240005212
// MI455X (gfx1250) — compile-verified
//
#include <hip/hip_runtime.h>
#include <hip/hip_bf16.h>
#include <math.h>

// Problem constants (match reference).
#define BB 128
#define TT 512
#define KK 1024
#define QQ 1024

typedef __attribute__((ext_vector_type(2))) float v2f;
typedef __attribute__((ext_vector_type(4))) float v4f;
typedef __attribute__((ext_vector_type(8))) float v8f;

// ---------------------------------------------------------------------------
// Kernel 1: mids[b,k] = sum_q W[k,q] * query[b,q]
// GEMM  M=128 (batch), N=1024 (k), inner=1024 (q), fp32 via WMMA 16x16x4.
// One wave (32 threads) per 16x16 output tile; 256 WMMAs per wave.
//
// fp32 A 16x4 layout (ISA 7.12.2): lanes 0-15 hold M=lane, VGPR{0,1}=K{0,1};
// lanes 16-31 hold M=lane-16, VGPR{0,1}=K{2,3}.  B 4x16 mirrors: lane group
// selects K pair, lane%16 selects N column.  D 16x16: VGPR r -> row
// (lane<16 ? r : 8+r), col = lane%16.
// ---------------------------------------------------------------------------
__global__ __launch_bounds__(32)
void mids_wmma_kernel(const float* __restrict__ query,
                      const float* __restrict__ W,
                      float* __restrict__ mids)
{
    const int tile = blockIdx.x;        // 0..511
    const int tm   = tile >> 6;         // 0..7   batch tile
    const int tn   = tile & 63;         // 0..63  k tile
    const int lane = threadIdx.x;       // 0..31
    const int half = lane >> 4;         // 0 or 1 (K-pair group)
    const int l16  = lane & 15;

    const int m = tm * 16 + l16;        // batch row for A fragment
    const int n = tn * 16 + l16;        // k column for B fragment

    const float* __restrict__ qrow = query + (size_t)m * QQ;  // A: query row m
    const float* __restrict__ wrow = W     + (size_t)n * QQ;  // B col n of W^T = row n of W

    v8f acc = {};
    #pragma unroll 4
    for (int q0 = 0; q0 < QQ; q0 += 4) {
        // A fragment: K = q0 + 2*half + {0,1}
        v2f a = *(const v2f*)(qrow + q0 + 2 * half);
        // B fragment: rows K = q0 + 2*half + {0,1}, column n
        v2f b = *(const v2f*)(wrow + q0 + 2 * half);
        acc = __builtin_amdgcn_wmma_f32_16x16x4_f32(
            /*neg_a=*/false, a, /*neg_b=*/false, b,
            /*c_mod=*/(short)0, acc, /*reuse_a=*/false, /*reuse_b=*/false);
    }

    // Scatter D tile.
    const int row_base = tm * 16 + half * 8;
    const int col      = tn * 16 + l16;
    #pragma unroll
    for (int r = 0; r < 8; ++r) {
        mids[(size_t)(row_base + r) * KK + col] = acc[r];
    }
}

// ---------------------------------------------------------------------------
// Kernel 2: scores[b,t] = tanh(dot(key[b,t,:], mids[b,:]) + bias)
// Bandwidth-bound streamer: grid = B * (T/64), 256 threads = 8 waves,
// each wave owns 8 consecutive t values.
//   - mids[b] (4 KB) is staged in LDS once per block, so the inner loop's
//     only global traffic is the key stream (ds_load for mids).
//   - key loads are non-temporal (TH=NT): 268 MB read-once stream > 192 MB
//     L2, so we keep it from thrashing the cache.
// ---------------------------------------------------------------------------
__global__ __launch_bounds__(256)
void scores_kernel(const float* __restrict__ key,
                   const float* __restrict__ mids,
                   const float* __restrict__ bias,
                   float* __restrict__ scores)
{
    __shared__ __align__(16) float smids[KK];   // 4 KB

    const int blk    = blockIdx.x;      // 0 .. B*8-1
    const int b      = blk >> 3;
    const int tchunk = blk & 7;         // which 64-t slice
    const int wave   = threadIdx.x >> 5;
    const int lane   = threadIdx.x & 31;

    const float bias0 = bias[0];
    const float* __restrict__ midsb = mids + (size_t)b * KK;

    // Stage mids[b] into LDS: 256 threads x float4 = 1024 floats.
    {
        const int k = threadIdx.x * 4;
        *(v4f*)(smids + k) = *(const v4f*)(midsb + k);
    }
    __syncthreads();

    #pragma unroll
    for (int i = 0; i < 8; ++i) {
        const int t = tchunk * 64 + wave * 8 + i;
        const float* __restrict__ krow = key + ((size_t)b * TT + t) * KK;

        float a0 = 0.f, a1 = 0.f, a2 = 0.f, a3 = 0.f;
        #pragma unroll
        for (int k0 = 0; k0 < KK; k0 += 128) {
            const int k = k0 + lane * 4;
            v4f kv = __builtin_nontemporal_load((const v4f*)(krow + k));
            v4f mv = *(const v4f*)(smids + k);
            a0 = fmaf(kv.x, mv.x, a0);
            a1 = fmaf(kv.y, mv.y, a1);
            a2 = fmaf(kv.z, mv.z, a2);
            a3 = fmaf(kv.w, mv.w, a3);
        }
        float s = (a0 + a1) + (a2 + a3);
        // wave32 reduce
        #pragma unroll
        for (int off = 16; off > 0; off >>= 1)
            s += __shfl_down(s, off, 32);
        if (lane == 0)
            scores[(size_t)b * TT + t] = tanhf(s + bias0);
    }
}

// ---------------------------------------------------------------------------
// Kernel 3: in-place masked softmax over T per batch row (on d_out).
// One 256-thread block per b; each thread holds 2 of the 512 elements.
// ---------------------------------------------------------------------------
__global__ __launch_bounds__(256)
void softmax_kernel(const float* __restrict__ mask,
                    float* __restrict__ out)
{
    __shared__ float sred[8];
    const int b    = blockIdx.x;
    const int tid  = threadIdx.x;       // 0..255
    const int lane = tid & 31;
    const int wave = tid >> 5;

    float* __restrict__ row        = out  + (size_t)b * TT;
    const float* __restrict__ mrow = mask + (size_t)b * TT;

    float v0 = row[tid];
    float v1 = row[tid + 256];

    // ---- block max (unmasked, as in reference) ----
    float mx = fmaxf(v0, v1);
    #pragma unroll
    for (int off = 16; off > 0; off >>= 1)
        mx = fmaxf(mx, __shfl_down(mx, off, 32));
    if (lane == 0) sred[wave] = mx;
    __syncthreads();
    if (wave == 0) {
        float m = sred[lane & 7];
        #pragma unroll
        for (int off = 4; off > 0; off >>= 1)
            m = fmaxf(m, __shfl_down(m, off, 32));
        if (lane == 0) sred[0] = m;
    }
    __syncthreads();
    mx = sred[0];
    __syncthreads();

    // ---- exp, mask, block sum ----
    float e0 = __expf(v0 - mx) * mrow[tid];
    float e1 = __expf(v1 - mx) * mrow[tid + 256];
    float s  = e0 + e1;
    #pragma unroll
    for (int off = 16; off > 0; off >>= 1)
        s += __shfl_down(s, off, 32);
    if (lane == 0) sred[wave] = s;
    __syncthreads();
    if (wave == 0) {
        float t = sred[lane & 7];
        #pragma unroll
        for (int off = 4; off > 0; off >>= 1)
            t += __shfl_down(t, off, 32);
        if (lane == 0) sred[0] = t;
    }
    __syncthreads();
    const float inv = 1.0f / sred[0];

    row[tid]       = e0 * inv;
    row[tid + 256] = e1 * inv;
}

// ---------------------------------------------------------------------------
// Launcher. Inputs (setup_inputs order): query[B,Q], key[B,T,K], mask[B,T],
// W[K,Q], bias[1].  Output: attn_weights [B,T] fp32.
// d_ws: mids [B,K] fp32 = 512 KB.
// ---------------------------------------------------------------------------
extern "C" void kernel_launch(void* const* d_in, const int* in_sizes, int n_in,
                              void* d_out, int out_size, void* d_ws, size_t ws_size,
                              hipStream_t stream)
{
    const float* query = (const float*)d_in[0];
    const float* key   = (const float*)d_in[1];
    const float* mask  = (const float*)d_in[2];
    const float* W     = (const float*)d_in[3];
    const float* bias  = (const float*)d_in[4];
    float* out  = (float*)d_out;
    float* mids = (float*)d_ws;          // B*K floats = 512 KB

    // 1) mids = query @ W^T via fp32 WMMA: (128/16)*(1024/16) = 512 tiles,
    //    one wave per tile.
    mids_wmma_kernel<<<512, 32, 0, stream>>>(query, W, mids);

    // 2) scores = tanh(key . mids + bias) -> d_out. 1024 blocks stream 268 MB
    //    non-temporally with mids served from LDS.
    scores_kernel<<<BB * (TT / 64), 256, 0, stream>>>(key, mids, bias, out);

    // 3) masked softmax over T, in place on d_out.
    softmax_kernel<<<BB, 256, 0, stream>>>(mask, out);
}